// GlobalAlignLoss_49615462203609
// MI455X (gfx1250) — compile-verified
//
#include <hip/hip_runtime.h>

// Chamfer-style global alignment loss for MI455X (gfx1250, wave32).
// Dot products via V_WMMA_F32_16X16X4_F32 (K=3 padded to 4); 2 tiles/iter
// with both global_load_b96 issued up front so load latency overlaps the
// first tile's WMMA+min work. Min-reductions in registers + ds_swizzle;
// cross-block merge via monotone-keyed u32 global atomic min.

typedef float v2f __attribute__((ext_vector_type(2)));
typedef float v8f __attribute__((ext_vector_type(8)));

#define B_ 4
#define N_ 8192
#define M_ 8192

// ds_swizzle immediate: group-of-32, xor_mask<<10 | and_mask(0x1f)
#define SWZ_XOR(x, mask)                                             \
  __uint_as_float((unsigned)__builtin_amdgcn_ds_swizzle(             \
      (int)__float_as_uint(x), ((mask) << 10) | 0x1f))

// Monotone float -> uint key (order-preserving under unsigned compare).
__device__ __forceinline__ unsigned fkey(float f) {
  unsigned u = __float_as_uint(f);
  return (u & 0x80000000u) ? ~u : (u | 0x80000000u);
}
__device__ __forceinline__ float kinv(unsigned k) {
  unsigned u = (k & 0x80000000u) ? (k ^ 0x80000000u) : ~k;
  return __uint_as_float(u);
}

__global__ __launch_bounds__(256) void init_kernel(unsigned* __restrict__ mins,
                                                   int n, float* __restrict__ out) {
  int i = blockIdx.x * blockDim.x + threadIdx.x;
  if (i < n) mins[i] = 0xFF800000u;  // fkey(+inf)
  if (i == 0) out[0] = 0.0f;
}

__global__ __launch_bounds__(256) void chamfer_min_kernel(
    const float* __restrict__ preds,   // [B, M, 3]  (y)
    const float* __restrict__ gts,     // [B, N, 3]  (x)
    unsigned* __restrict__ rowmin,     // [B, N] keys: min over j of P[b,i,j]
    unsigned* __restrict__ colmin) {   // [B, M] keys: min over i of P[b,i,j]
  const int ntiles_n = N_ / 16;
  const int b    = blockIdx.x / ntiles_n;
  const int i0   = (blockIdx.x % ntiles_n) * 16;
  const int lane = threadIdx.x & 31;
  const int wave = threadIdx.x >> 5;
  const int half = lane >> 4;   // fragment half: 0 -> K={0,1}, 1 -> K={2,3}
  const int l16  = lane & 15;

  // ---- A fragment: 16x4 f32 tile of gts rows i0..i0+15, K=3 zero-padded ----
  const float* xr = gts + ((size_t)b * N_ + (size_t)(i0 + l16)) * 3;
  const float x0 = xr[0], x1 = xr[1], x2 = xr[2];
  const float rx_own = x0 * x0 + x1 * x1 + x2 * x2;
  v2f a;
  a.x = half ? x2   : x0;   // v_cndmask, no branch
  a.y = half ? 0.0f : x1;

  // rx for the 8 rows this lane's C accumulators represent (rows v + 8*half).
  float rxv[8];
#pragma unroll
  for (int v = 0; v < 8; ++v)
    rxv[v] = __shfl(rx_own, v + 8 * half, 32);  // one-time setup

  float rmin[8];
#pragma unroll
  for (int v = 0; v < 8; ++v) rmin[v] = __builtin_inff();

  const size_t bM = (size_t)b * M_;

  // ---- sweep column tiles of preds: 8 waves x 2 tiles/iter cover 512 ----
  for (int jt = wave * 2; jt < M_ / 16; jt += 16) {
    const int col0 = jt * 16 + l16;
    const int col1 = col0 + 16;

    // Issue both tile loads up front (two global_load_b96 in flight).
    const float* yr0 = preds + (bM + (size_t)col0) * 3;
    const float* yr1 = preds + (bM + (size_t)col1) * 3;
    const float p0 = yr0[0], p1 = yr0[1], p2 = yr0[2];
    const float q0 = yr1[0], q1 = yr1[1], q2 = yr1[2];

    // Tile 0
    const float ry0 = p0 * p0 + p1 * p1 + p2 * p2;
    v2f b0;
    b0.x = half ? p2   : p0;
    b0.y = half ? 0.0f : p1;
    v8f c0 = {};
    c0 = __builtin_amdgcn_wmma_f32_16x16x4_f32(
        false, a, false, b0, (short)0, c0, false, false);

    // Tile 1
    const float ry1 = q0 * q0 + q1 * q1 + q2 * q2;
    v2f b1;
    b1.x = half ? q2   : q0;
    b1.y = half ? 0.0f : q1;
    v8f c1 = {};
    c1 = __builtin_amdgcn_wmma_f32_16x16x4_f32(
        false, a, false, b1, (short)0, c1, false, false);

    float cmin0 = __builtin_inff();
    float cmin1 = __builtin_inff();
#pragma unroll
    for (int v = 0; v < 8; ++v) {
      float P0 = rxv[v] + ry0 - 2.0f * c0[v];
      float P1 = rxv[v] + ry1 - 2.0f * c1[v];
      rmin[v] = fminf(rmin[v], fminf(P0, P1));
      cmin0   = fminf(cmin0, P0);
      cmin1   = fminf(cmin1, P1);
    }
    // column min over all 16 strip rows: merge the two 8-row halves
    cmin0 = fminf(cmin0, SWZ_XOR(cmin0, 0x10));
    cmin1 = fminf(cmin1, SWZ_XOR(cmin1, 0x10));
    if (half == 0) {   // one exec-mask region, two atomics
      atomicMin(&colmin[bM + (size_t)col0], fkey(cmin0));
      atomicMin(&colmin[bM + (size_t)col1], fkey(cmin1));
    }
  }

  // ---- row mins: butterfly over the 16 lanes of each half ----
#pragma unroll
  for (int v = 0; v < 8; ++v) {
    float r = rmin[v];
    r = fminf(r, SWZ_XOR(r, 0x01));
    r = fminf(r, SWZ_XOR(r, 0x02));
    r = fminf(r, SWZ_XOR(r, 0x04));
    r = fminf(r, SWZ_XOR(r, 0x08));
    rmin[v] = r;
  }
  if (l16 == 0) {
#pragma unroll
    for (int v = 0; v < 8; ++v)
      atomicMin(&rowmin[(size_t)b * N_ + (size_t)(i0 + v + 8 * half)],
                fkey(rmin[v]));
  }
}

__global__ __launch_bounds__(256) void finalize_kernel(
    const unsigned* __restrict__ mins, int n,
    const int* __restrict__ cptr, float* __restrict__ out) {
  __shared__ float sm[256];
  const float cf = (float)cptr[0];
  float s = 0.0f;
  for (int i = blockIdx.x * blockDim.x + threadIdx.x; i < n;
       i += gridDim.x * blockDim.x) {
    float v = kinv(mins[i]);
    s += (v < cf) ? (0.5f * v * v) : (cf * v - 0.5f * cf * cf);
  }
  sm[threadIdx.x] = s;
  __syncthreads();
  for (int off = 128; off > 0; off >>= 1) {
    if (threadIdx.x < off) sm[threadIdx.x] += sm[threadIdx.x + off];
    __syncthreads();
  }
  if (threadIdx.x == 0) atomicAdd(out, sm[0]);
}

extern "C" void kernel_launch(void* const* d_in, const int* in_sizes, int n_in,
                              void* d_out, int out_size, void* d_ws, size_t ws_size,
                              hipStream_t stream) {
  const float* preds = (const float*)d_in[0];   // [B, M, 3]
  const float* gts   = (const float*)d_in[1];   // [B, N, 3]
  const int*   c     = (const int*)d_in[2];     // scalar
  float* out = (float*)d_out;

  unsigned* rowmin = (unsigned*)d_ws;                  // B*N keys
  unsigned* colmin = rowmin + (size_t)B_ * N_;         // B*M keys
  const int total  = B_ * N_ + B_ * M_;

  init_kernel<<<(total + 255) / 256, 256, 0, stream>>>(rowmin, total, out);
  chamfer_min_kernel<<<B_ * (N_ / 16), 256, 0, stream>>>(preds, gts, rowmin, colmin);
  finalize_kernel<<<256, 256, 0, stream>>>(rowmin, total, c, out);
}